// Discriminator_75557064671746
// MI455X (gfx1250) — compile-verified
//
#include <hip/hip_runtime.h>
#include <hip/hip_bf16.h>
#include <math.h>

typedef __attribute__((ext_vector_type(2))) float v2f;
typedef __attribute__((ext_vector_type(8))) float v8f;

__device__ __forceinline__ float sigmoid_f(float z) {
    return 1.0f / (1.0f + expf(-z));
}

// ---------------------------------------------------------------------------
// init: deg = 1.0 (self-loop contribution), s1 = s2 = 0
// ---------------------------------------------------------------------------
__global__ void k_init(float* __restrict__ deg, float* __restrict__ s1,
                       float* __restrict__ s2, int n) {
    int i = blockIdx.x * blockDim.x + threadIdx.x;
    if (i < n) { deg[i] = 1.0f; s1[i] = 0.0f; s2[i] = 0.0f; }
}

// ---------------------------------------------------------------------------
// degree: deg[dst[e]] += 1 over the 1.6M real edges
// ---------------------------------------------------------------------------
__global__ void k_degree(const int* __restrict__ dst, float* __restrict__ deg,
                         int E) {
    int i = blockIdx.x * blockDim.x + threadIdx.x;
    if (i < E) atomicAdd(&deg[dst[i]], 1.0f);
}

// ---------------------------------------------------------------------------
// dinv = rsqrt(deg) in place (deg >= 1 always due to self-loops)
// ---------------------------------------------------------------------------
__global__ void k_rsqrt(float* __restrict__ deg, int n) {
    int i = blockIdx.x * blockDim.x + threadIdx.x;
    if (i < n) {
        float d = deg[i];
        deg[i] = (d > 0.0f) ? rsqrtf(d) : 0.0f;
    }
}

// ---------------------------------------------------------------------------
// scalar edge aggregation (shared by both layers):
//   out[d] += val[s] * dinv[s] * dinv[d]  over E edges + N self-loops
// ---------------------------------------------------------------------------
__global__ void k_agg(const int* __restrict__ src, const int* __restrict__ dst,
                      const float* __restrict__ val,
                      const float* __restrict__ dinv,
                      float* __restrict__ out, int E, int n) {
    int i = blockIdx.x * blockDim.x + threadIdx.x;
    if (i >= E + n) return;
    int s, d;
    if (i < E) { s = src[i]; d = dst[i]; }
    else       { s = d = i - E; }            // self-loop
    atomicAdd(&out[d], val[s] * dinv[s] * dinv[d]);
}

// ---------------------------------------------------------------------------
// Fused layer-1 activation + layer-2 GEMV via WMMA:
//   h2[i] = sum_j sigmoid(s1[i]*W1[j] + b1[j]) * W2[j],  j = 0..34
// One wave handles a 16-node tile. D = A(16x4) x B(4x16) + C, 9 K-chunks.
//   A: only row M=0 populated with W2[k] (zeros past k=34 -> padding exact).
//   B: B[k][m] = sigmoid(s1[node m]*W1[k] + b1[k]), generated in registers.
//   D: row M=0 = h2 for the 16 nodes, in acc[0], lanes 0..15.
// ISA layouts (7.12.2): A lane L<16 -> M=L, v0=K0,v1=K1; lane>=16 -> K2,K3.
//                       B v0 = rows K0(lo)/K2(hi), v1 = rows K1(lo)/K3(hi).
// ---------------------------------------------------------------------------
__global__ void k_transform_wmma(const float* __restrict__ s1,
                                 const float* __restrict__ W1,
                                 const float* __restrict__ b1,
                                 const float* __restrict__ W2,
                                 float* __restrict__ h2, int n) {
    const int lane = threadIdx.x & 31;
    const int wave = (blockIdx.x * blockDim.x + threadIdx.x) >> 5;
    const int base = wave * 16;
    if (base >= n) return;                 // wave-uniform: EXEC stays all-1s

    const int  m    = lane & 15;           // column (node) index for B/D
    const int  koff = (lane >> 4) << 1;    // 0 for lanes 0-15, 2 for 16-31
    int node = base + m;
    int nodeC = (node < n) ? node : (n - 1);
    float s = s1[nodeC];

    v8f acc = {};
    #pragma unroll
    for (int c = 0; c < 9; ++c) {          // K = 36 (35 real + 1 zero pad)
        int k0 = 4 * c + koff;
        int k1 = k0 + 1;
        float w1a = (k0 < 35) ? W1[k0] : 0.0f;
        float w1b = (k1 < 35) ? W1[k1] : 0.0f;
        float ba  = (k0 < 35) ? b1[k0] : 0.0f;
        float bb  = (k1 < 35) ? b1[k1] : 0.0f;

        v2f Bm;                            // x1 values for this wave's tile
        Bm.x = sigmoid_f(fmaf(s, w1a, ba));
        Bm.y = sigmoid_f(fmaf(s, w1b, bb));

        v2f Am;                            // W2 in row M=0 only; 0 when k>=35
        Am.x = (m == 0 && k0 < 35) ? W2[k0] : 0.0f;
        Am.y = (m == 0 && k1 < 35) ? W2[k1] : 0.0f;

        acc = __builtin_amdgcn_wmma_f32_16x16x4_f32(
            /*neg_a=*/false, Am, /*neg_b=*/false, Bm,
            /*c_mod=*/(short)0, acc, /*reuse_a=*/false, /*reuse_b=*/false);
    }
    if (lane < 16 && node < n) h2[node] = acc[0];   // D row M=0, N=lane
}

// ---------------------------------------------------------------------------
// finalize: out[i] = sigmoid(s2[i] + b2)
// ---------------------------------------------------------------------------
__global__ void k_final(const float* __restrict__ s2,
                        const float* __restrict__ b2,
                        float* __restrict__ out, int n) {
    int i = blockIdx.x * blockDim.x + threadIdx.x;
    if (i < n) out[i] = sigmoid_f(s2[i] + b2[0]);
}

extern "C" void kernel_launch(void* const* d_in, const int* in_sizes, int n_in,
                              void* d_out, int out_size, void* d_ws, size_t ws_size,
                              hipStream_t stream) {
    const float* x    = (const float*)d_in[0];
    const int*   eidx = (const int*)  d_in[1];   // [2, E] row-major
    const float* W1   = (const float*)d_in[2];
    const float* b1   = (const float*)d_in[3];
    const float* W2   = (const float*)d_in[4];
    const float* b2   = (const float*)d_in[5];

    const int n = in_sizes[0];                   // 100000 nodes
    const int E = in_sizes[1] / 2;               // 1600000 edges
    const int* src = eidx;
    const int* dst = eidx + E;

    float* deg = (float*)d_ws;                   // becomes dinv in place
    float* s1  = deg + n;
    float* h2  = s1 + n;
    float* s2  = h2 + n;
    float* out = (float*)d_out;

    const int B = 256;
    const int nb_n = (n + B - 1) / B;
    const int nb_e = (E + B - 1) / B;
    const int nb_a = (E + n + B - 1) / B;

    k_init  <<<nb_n, B, 0, stream>>>(deg, s1, s2, n);
    k_degree<<<nb_e, B, 0, stream>>>(dst, deg, E);
    k_rsqrt <<<nb_n, B, 0, stream>>>(deg, n);
    k_agg   <<<nb_a, B, 0, stream>>>(src, dst, x, deg, s1, E, n);

    const int tiles = (n + 15) / 16;             // 16 nodes per wave
    const int wavesPerBlock = B / 32;            // 8 waves (wave32)
    const int nb_w = (tiles + wavesPerBlock - 1) / wavesPerBlock;
    k_transform_wmma<<<nb_w, B, 0, stream>>>(s1, W1, b1, W2, h2, n);

    k_agg   <<<nb_a, B, 0, stream>>>(src, dst, h2, deg, s2, E, n);
    k_final <<<nb_n, B, 0, stream>>>(s2, b2, out, n);
}